// VAE_19834158973316
// MI455X (gfx1250) — compile-verified
//
#include <hip/hip_runtime.h>
#include <math.h>

typedef float v2f __attribute__((ext_vector_type(2)));
typedef float v8f __attribute__((ext_vector_type(8)));

#define GDIM 64
#define ZDIM 32

// ---------------- degree / norm ----------------

__global__ void deg_kernel(const int* __restrict__ dst, float* __restrict__ deg, int E) {
    int i = blockIdx.x * blockDim.x + threadIdx.x;
    if (i < E) unsafeAtomicAdd(&deg[dst[i]], 1.0f);
}

__global__ void norm_kernel(float* __restrict__ deg, int n) {
    int i = blockIdx.x * blockDim.x + threadIdx.x;
    if (i < n) {
        float d = deg[i];
        d = d < 1.0f ? 1.0f : d;
        deg[i] = rsqrtf(d);
    }
}

// ---------------- fused scale+gather+scatter ----------------
// one thread per (edge, 4-float chunk):
//   agg[dst[e], c*4..c*4+3] += h[src[e], c*4..c*4+3] * norm[src[e]]

template <int D>
__global__ void scatter_kernel(const float* __restrict__ h,
                               const float* __restrict__ norm,
                               const int* __restrict__ src,
                               const int* __restrict__ dst,
                               float* __restrict__ agg, int E) {
    const int CH = D / 4;
    int gid = blockIdx.x * blockDim.x + threadIdx.x;
    if (gid >= E * CH) return;
    int e = gid / CH;
    int c = gid % CH;
    int s = src[e];
    int t = dst[e];
    float ns = norm[s];
    const float4 v = reinterpret_cast<const float4*>(h + (size_t)s * D)[c];
    float* ap = agg + (size_t)t * D + c * 4;
    unsafeAtomicAdd(ap + 0, v.x * ns);
    unsafeAtomicAdd(ap + 1, v.y * ns);
    unsafeAtomicAdd(ap + 2, v.z * ns);
    unsafeAtomicAdd(ap + 3, v.w * ns);
}

// ---------------- WMMA node-apply: out = act((agg*norm) @ W^T + b) ----------------
// One wave handles a 16-node row tile; V_WMMA_F32_16X16X4_F32, K-loop in steps of 4.
// A 16x4 f32 layout: lanes 0-15 hold K={0,1}, lanes 16-31 hold K={2,3} (2 VGPRs).
// B 4x16 f32 layout mirrors A: lane holds B[kbase+{0,1}][col], col = lane%16.
// C/D 16x16 f32: VGPR j -> row j (lanes 0-15) / row j+8 (lanes 16-31), col = lane%16.

enum { ACT_IDENT = 0, ACT_RELU = 1, ACT_SIGMOID = 2 };

template <int K, int O, int ACT>
__global__ void apply_kernel(const float* __restrict__ agg,
                             const float* __restrict__ norm,
                             const float* __restrict__ W,
                             const float* __restrict__ bias,
                             float* __restrict__ out, int ntiles) {
    int wave = blockIdx.x * (blockDim.x >> 5) + (threadIdx.x >> 5);
    if (wave >= ntiles) return;   // wave-uniform: EXEC stays all-ones for WMMA
    int lane = threadIdx.x & 31;
    int mrow = lane & 15;
    int hi   = lane >> 4;         // 0: K pair {0,1}; 1: K pair {2,3}

    int row = wave * 16 + mrow;
    float nr = norm[row];

    // preload A fragments for all K steps (agg row is L2-resident)
    v2f afrag[K / 4];
    const float* arow = agg + (size_t)row * K;
#pragma unroll
    for (int kk = 0; kk < K / 4; ++kk) {
        v2f a = *reinterpret_cast<const v2f*>(arow + kk * 4 + hi * 2);
        afrag[kk] = a * nr;
    }

#pragma unroll
    for (int ot = 0; ot < O / 16; ++ot) {
        int col = ot * 16 + mrow;
        const float* wrow = W + (size_t)col * K;   // B[k][col] = W[col][k]
        v8f c = {0.f, 0.f, 0.f, 0.f, 0.f, 0.f, 0.f, 0.f};
#pragma unroll
        for (int kk = 0; kk < K / 4; ++kk) {
            v2f bf = *reinterpret_cast<const v2f*>(wrow + kk * 4 + hi * 2);
            c = __builtin_amdgcn_wmma_f32_16x16x4_f32(
                    false, afrag[kk], false, bf, (short)0, c, false, false);
        }
        float bv = bias[col];
#pragma unroll
        for (int j = 0; j < 8; ++j) {
            int r = wave * 16 + j + hi * 8;
            float v = c[j] + bv;
            if (ACT == ACT_RELU)         v = v > 0.f ? v : 0.f;
            else if (ACT == ACT_SIGMOID) v = 1.f / (1.f + __expf(-v));
            out[(size_t)r * O + col] = v;
        }
    }
}

// ---------------- reparameterization ----------------

__global__ void z_kernel(const float* __restrict__ eps,
                         const float* __restrict__ mu,
                         const float* __restrict__ logvar,
                         float* __restrict__ z, int n) {
    int i = blockIdx.x * blockDim.x + threadIdx.x;
    if (i < n) z[i] = eps[i] * __expf(0.5f * logvar[i]) + mu[i];
}

// ---------------- driver ----------------

extern "C" void kernel_launch(void* const* d_in, const int* in_sizes, int n_in,
                              void* d_out, int out_size, void* d_ws, size_t ws_size,
                              hipStream_t stream) {
    const float* x    = (const float*)d_in[0];
    const int*   esrc = (const int*)d_in[1];
    const int*   edst = (const int*)d_in[2];
    const float* eps  = (const float*)d_in[3];
    const float* W1  = (const float*)d_in[4];  const float* b1  = (const float*)d_in[5];
    const float* W2  = (const float*)d_in[6];  const float* b2  = (const float*)d_in[7];
    const float* W31 = (const float*)d_in[8];  const float* b31 = (const float*)d_in[9];
    const float* W32 = (const float*)d_in[10]; const float* b32 = (const float*)d_in[11];
    const float* W4  = (const float*)d_in[12]; const float* b4  = (const float*)d_in[13];
    const float* W5  = (const float*)d_in[14]; const float* b5  = (const float*)d_in[15];
    const float* W6  = (const float*)d_in[16]; const float* b6  = (const float*)d_in[17];

    const int N = in_sizes[0] / GDIM;   // 100000
    const int E = in_sizes[1];          // 1600000

    float* out    = (float*)d_out;
    float* mu     = out + (size_t)N * GDIM;
    float* logvar = mu  + (size_t)N * ZDIM;

    // workspace: norm[N] | agg[N*64] | h[N*64]   (~52 MB)
    float* norm = (float*)d_ws;
    float* agg  = norm + (size_t)N;
    float* h    = agg  + (size_t)N * 64;

    const int TPB = 256;
    const int ntiles = N / 16;                      // 6250 (N divisible by 16)
    const dim3 ablk(128);                           // 4 waves per block
    const int agrid = (ntiles + 3) / 4;

    // degrees -> norm = rsqrt(max(deg,1))
    hipMemsetAsync(norm, 0, (size_t)N * sizeof(float), stream);
    deg_kernel<<<(E + TPB - 1) / TPB, TPB, 0, stream>>>(edst, norm, E);
    norm_kernel<<<(N + TPB - 1) / TPB, TPB, 0, stream>>>(norm, N);

    auto scat64 = [&](const float* hin) {
        hipMemsetAsync(agg, 0, (size_t)N * 64 * sizeof(float), stream);
        int t = E * 16;
        scatter_kernel<64><<<(t + TPB - 1) / TPB, TPB, 0, stream>>>(
            hin, norm, esrc, edst, agg, E);
    };

    // layer 1: x -> h (relu)
    scat64(x);
    apply_kernel<64, 64, ACT_RELU><<<agrid, ablk, 0, stream>>>(agg, norm, W1, b1, h, ntiles);

    // layer 2: h -> h (relu)
    scat64(h);
    apply_kernel<64, 64, ACT_RELU><<<agrid, ablk, 0, stream>>>(agg, norm, W2, b2, h, ntiles);

    // layer 3: shared message passing, two heads (mu, log_var), identity
    scat64(h);
    apply_kernel<64, 32, ACT_IDENT><<<agrid, ablk, 0, stream>>>(agg, norm, W31, b31, mu, ntiles);
    apply_kernel<64, 32, ACT_IDENT><<<agrid, ablk, 0, stream>>>(agg, norm, W32, b32, logvar, ntiles);

    // z = eps * exp(0.5*log_var) + mu  (stored in h, width 32)
    z_kernel<<<((size_t)N * ZDIM + TPB - 1) / TPB, TPB, 0, stream>>>(eps, mu, logvar, h, N * ZDIM);

    // layer 4: z(32) -> h(64) (relu)
    hipMemsetAsync(agg, 0, (size_t)N * 32 * sizeof(float), stream);
    {
        int t = E * 8;
        scatter_kernel<32><<<(t + TPB - 1) / TPB, TPB, 0, stream>>>(
            h, norm, esrc, edst, agg, E);
    }
    apply_kernel<32, 64, ACT_RELU><<<agrid, ablk, 0, stream>>>(agg, norm, W4, b4, h, ntiles);

    // layer 5: h -> h (relu)
    scat64(h);
    apply_kernel<64, 64, ACT_RELU><<<agrid, ablk, 0, stream>>>(agg, norm, W5, b5, h, ntiles);

    // layer 6: h -> recon (sigmoid)
    scat64(h);
    apply_kernel<64, 64, ACT_SIGMOID><<<agrid, ablk, 0, stream>>>(agg, norm, W6, b6, out, ntiles);
}